// CircleLoss_17884243820936
// MI455X (gfx1250) — compile-verified
//
#include <hip/hip_runtime.h>
#include <math.h>

#define MARG  0.25f
#define GAMMA 256.0f
#define BN    8192
#define DIM   256

typedef __attribute__((ext_vector_type(2))) float v2f;
typedef __attribute__((ext_vector_type(8))) float v8f;

// ---- online logsumexp helpers (sentinel -1e30 keeps everything NaN-free) ----
__device__ __forceinline__ void lse_update(float& m, float& s, float v) {
    float M = fmaxf(m, v);
    s = s * __expf(m - M) + __expf(v - M);
    m = M;
}
__device__ __forceinline__ void lse_merge(float& m, float& s, float m2, float s2) {
    float M = fmaxf(m, m2);
    s = s * __expf(m - M) + s2 * __expf(m2 - M);
    m = M;
}

// ---------------- Kernel 1: reciprocal row norms ----------------
__global__ void __launch_bounds__(256)
rnorm_kernel(const float* __restrict__ emb, float* __restrict__ rn) {
    const int row = blockIdx.x;
    float v = emb[row * DIM + threadIdx.x];
    float ss = v * v;
    #pragma unroll
    for (int off = 16; off > 0; off >>= 1) ss += __shfl_xor(ss, off, 32);
    __shared__ float lds[8];
    if ((threadIdx.x & 31) == 0) lds[threadIdx.x >> 5] = ss;
    __syncthreads();
    if (threadIdx.x == 0) {
        float tot = 0.f;
        #pragma unroll
        for (int i = 0; i < 8; ++i) tot += lds[i];
        rn[row] = 1.0f / fmaxf(sqrtf(tot), 1e-12f);
    }
}

// ---- Kernel 2: fused Gram-matrix WMMA + masked online logsumexp ----
// Grid (64,64): 128x128 output tile per block, 8 waves arranged 4x2, each
// wave owns a 32x64 strip = 2 A-frags x 4 B-frags = 8 accumulators.
// Inner loop: 6 loads per 8 V_WMMA_F32_16X16X4_F32 (A reused 4x, B 2x).
__global__ void __launch_bounds__(256)
circle_tile_kernel(const float* __restrict__ emb,
                   const int*   __restrict__ labels,
                   const float* __restrict__ rn,
                   float*       __restrict__ partials) {
    const int w    = threadIdx.x >> 5;
    const int lane = threadIdx.x & 31;
    const int half = lane >> 4;
    const int l15  = lane & 15;
    const int wr   = w >> 1;   // 0..3 row group
    const int wc   = w & 1;    // 0..1 col group
    const int rowBase = blockIdx.y * 128 + wr * 32;
    const int colBase = blockIdx.x * 128 + wc * 64;

    // A fragment rows for this lane, B fragment rows for the 4 column tiles.
    // Both fragments use the same per-lane address formula (Gram matrix!).
    const float* arow0 = emb + (size_t)(rowBase + l15) * DIM + 2 * half;
    const float* arow1 = arow0 + (size_t)16 * DIM;
    const float* bbase = emb + (size_t)(colBase + l15) * DIM + 2 * half;

    v8f zero = {};
    v8f c[2][4];
    #pragma unroll
    for (int i = 0; i < 2; ++i)
        #pragma unroll
        for (int t = 0; t < 4; ++t) c[i][t] = zero;

    for (int k0 = 0; k0 < DIM; k0 += 4) {
        v2f a0 = *(const v2f*)(arow0 + k0);
        v2f a1 = *(const v2f*)(arow1 + k0);
        #pragma unroll
        for (int t = 0; t < 4; ++t) {
            v2f b = *(const v2f*)(bbase + (size_t)t * 16 * DIM + k0);
            c[0][t] = __builtin_amdgcn_wmma_f32_16x16x4_f32(
                false, a0, false, b, (short)0, c[0][t], false, false);
            c[1][t] = __builtin_amdgcn_wmma_f32_16x16x4_f32(
                false, a1, false, b, (short)0, c[1][t], false, false);
        }
    }

    // ---- epilogue: scale by reciprocal norms, mask, online LSE ----
    const int colLane0 = colBase + l15;
    int   lc[4];
    float rc[4];
    #pragma unroll
    for (int t = 0; t < 4; ++t) {
        lc[t] = labels[colLane0 + 16 * t];
        rc[t] = rn[colLane0 + 16 * t];
    }

    float mp = -1e30f, sp = 0.f, mn = -1e30f, sn = 0.f;
    #pragma unroll
    for (int i = 0; i < 2; ++i) {
        #pragma unroll
        for (int j = 0; j < 8; ++j) {
            // C/D layout: lanes 0-15 -> M=j, lanes 16-31 -> M=8+j ; N = l15
            const int   row = rowBase + i * 16 + half * 8 + j;
            const int   lr  = labels[row];
            const float rr  = rn[row];
            #pragma unroll
            for (int t = 0; t < 4; ++t) {
                const int   col = colLane0 + 16 * t;
                const float s   = c[i][t][j] * rr * rc[t];
                if (lr == lc[t]) {
                    if (row != col) {                      // positive pair
                        float ap = fmaxf(1.0f + MARG - s, 0.0f);
                        lse_update(mp, sp, -GAMMA * ap * (s - (1.0f - MARG)));
                    }
                } else {                                    // negative pair
                    float an = fmaxf(s + MARG, 0.0f);
                    lse_update(mn, sn, GAMMA * an * (s - MARG));
                }
            }
        }
    }

    // wave32 butterfly reduction
    #pragma unroll
    for (int off = 16; off > 0; off >>= 1) {
        lse_merge(mp, sp, __shfl_xor(mp, off, 32), __shfl_xor(sp, off, 32));
        lse_merge(mn, sn, __shfl_xor(mn, off, 32), __shfl_xor(sn, off, 32));
    }

    __shared__ float red[8][4];
    if (lane == 0) { red[w][0] = mp; red[w][1] = sp; red[w][2] = mn; red[w][3] = sn; }
    __syncthreads();
    if (threadIdx.x == 0) {
        for (int i = 1; i < 8; ++i) {
            lse_merge(mp, sp, red[i][0], red[i][1]);
            lse_merge(mn, sn, red[i][2], red[i][3]);
        }
        const int bid = blockIdx.y * gridDim.x + blockIdx.x;
        float4 r; r.x = mp; r.y = sp; r.z = mn; r.w = sn;
        ((float4*)partials)[bid] = r;
    }
}

// ---------------- Kernel 3: merge block partials -> scalar loss ----------------
__global__ void __launch_bounds__(256)
finalize_kernel(const float* __restrict__ partials, int nrec, float* __restrict__ out) {
    float mp = -1e30f, sp = 0.f, mn = -1e30f, sn = 0.f;
    for (int i = threadIdx.x; i < nrec; i += 256) {
        float4 r = ((const float4*)partials)[i];
        lse_merge(mp, sp, r.x, r.y);
        lse_merge(mn, sn, r.z, r.w);
    }
    #pragma unroll
    for (int off = 16; off > 0; off >>= 1) {
        lse_merge(mp, sp, __shfl_xor(mp, off, 32), __shfl_xor(sp, off, 32));
        lse_merge(mn, sn, __shfl_xor(mn, off, 32), __shfl_xor(sn, off, 32));
    }
    __shared__ float red[8][4];
    const int w = threadIdx.x >> 5;
    if ((threadIdx.x & 31) == 0) { red[w][0] = mp; red[w][1] = sp; red[w][2] = mn; red[w][3] = sn; }
    __syncthreads();
    if (threadIdx.x == 0) {
        for (int i = 1; i < 8; ++i) {
            lse_merge(mp, sp, red[i][0], red[i][1]);
            lse_merge(mn, sn, red[i][2], red[i][3]);
        }
        float lp  = (sp > 0.f) ? (mp + logf(sp)) : -1e30f;
        float ln  = (sn > 0.f) ? (mn + logf(sn)) : -1e30f;
        float tot = lp + ln;
        // overflow-safe softplus
        out[0] = fmaxf(tot, 0.f) + log1pf(__expf(-fabsf(tot)));
    }
}

extern "C" void kernel_launch(void* const* d_in, const int* in_sizes, int n_in,
                              void* d_out, int out_size, void* d_ws, size_t ws_size,
                              hipStream_t stream) {
    const float* emb    = (const float*)d_in[0];
    const int*   labels = (const int*)d_in[1];

    float* rn       = (float*)d_ws;                                       // 8192 f32 = 32 KB
    float* partials = (float*)((char*)d_ws + (size_t)BN * sizeof(float)); // 4096*16B = 64 KB

    rnorm_kernel<<<BN, 256, 0, stream>>>(emb, rn);

    dim3 grid(BN / 128, BN / 128);
    circle_tile_kernel<<<grid, 256, 0, stream>>>(emb, labels, rn, partials);

    finalize_kernel<<<1, 256, 0, stream>>>(partials, (BN / 128) * (BN / 128),
                                           (float*)d_out);
}